// TransitionDown_80513456931524
// MI455X (gfx1250) — compile-verified
//
#include <hip/hip_runtime.h>
#include <hip/hip_bf16.h>

#define B_    4
#define N_    8192
#define F_    64
#define S_    2048
#define K_    16
#define C_    128
#define CIN0  67
#define KPAD0 68
#define M_    (B_*S_*K_)   /* 131072 rows */
#define BN_EPS 1e-5f

typedef __attribute__((ext_vector_type(2))) float        v2f;
typedef __attribute__((ext_vector_type(8))) float        v8f;
typedef __attribute__((ext_vector_type(4))) unsigned int u32x4;
typedef __attribute__((ext_vector_type(4))) int          i32x4;
typedef __attribute__((ext_vector_type(8))) int          i32x8;

// ---------------------------------------------------------------------------
// 1) Farthest point sampling: one block per batch, min-dist resident in LDS,
//    block-tree argmax per step (tie-break: smallest index, matching argmax).
//    Writes new_xyz [B,S,3] directly into d_out.
// ---------------------------------------------------------------------------
__global__ void __launch_bounds__(1024)
fps_kernel(const float* __restrict__ xyz, float* __restrict__ new_xyz) {
    __shared__ float smind[N_];     // 32 KB
    __shared__ float sval[1024];
    __shared__ int   sidx[1024];
    const int b = blockIdx.x, tid = threadIdx.x;
    const float* px = xyz + (size_t)b * N_ * 3;
    float* pq = new_xyz + (size_t)b * S_ * 3;

    for (int i = tid; i < N_; i += 1024) smind[i] = 1e10f;
    __syncthreads();

    int cur = 0;
    for (int s = 0; s < S_; ++s) {
        const float cx = px[cur*3+0], cy = px[cur*3+1], cz = px[cur*3+2];
        if (tid == 0) { pq[s*3+0]=cx; pq[s*3+1]=cy; pq[s*3+2]=cz; }
        float bv = -1.0f; int bi = 0;
        for (int i = tid; i < N_; i += 1024) {
            float dx = px[i*3+0]-cx, dy = px[i*3+1]-cy, dz = px[i*3+2]-cz;
            float d  = dx*dx + dy*dy + dz*dz;
            float m  = fminf(smind[i], d);
            smind[i] = m;
            if (m > bv) { bv = m; bi = i; }   // keeps smallest idx per thread
        }
        sval[tid] = bv; sidx[tid] = bi;
        __syncthreads();
        for (int off = 512; off > 0; off >>= 1) {
            if (tid < off) {
                float ov = sval[tid+off]; int oi = sidx[tid+off];
                if (ov > sval[tid] || (ov == sval[tid] && oi < sidx[tid])) {
                    sval[tid] = ov; sidx[tid] = oi;
                }
            }
            __syncthreads();
        }
        cur = sidx[0];
        __syncthreads();
    }
}

// ---------------------------------------------------------------------------
// 2) kNN (K=16): one block per batch; point cloud (96KB) + queries (24KB)
//    staged in LDS; per-thread register top-16 with insertion sort.
// ---------------------------------------------------------------------------
__global__ void __launch_bounds__(256)
knn_kernel(const float* __restrict__ xyz, const float* __restrict__ new_xyz,
           int* __restrict__ knn) {
    extern __shared__ float sm[];
    float* sp = sm;              // N_*3
    float* sq = sm + N_*3;       // S_*3
    const int b = blockIdx.x, tid = threadIdx.x;
    const float* px = xyz + (size_t)b * N_ * 3;
    const float* pq = new_xyz + (size_t)b * S_ * 3;
    for (int i = tid; i < N_*3; i += 256) sp[i] = px[i];
    for (int i = tid; i < S_*3; i += 256) sq[i] = pq[i];
    __syncthreads();

    for (int q = tid; q < S_; q += 256) {
        const float qx = sq[q*3], qy = sq[q*3+1], qz = sq[q*3+2];
        float bd[K_]; int bi[K_];
#pragma unroll
        for (int t = 0; t < K_; ++t) { bd[t] = 3.4e38f; bi[t] = 0; }
        for (int j = 0; j < N_; ++j) {            // broadcast LDS reads
            float dx = sp[j*3]-qx, dy = sp[j*3+1]-qy, dz = sp[j*3+2]-qz;
            float d  = dx*dx + dy*dy + dz*dz;
            if (d < bd[K_-1]) {
                bd[K_-1] = d; bi[K_-1] = j;
#pragma unroll
                for (int t = K_-1; t >= 1; --t) {
                    if (bd[t] < bd[t-1]) {
                        float tv = bd[t]; bd[t] = bd[t-1]; bd[t-1] = tv;
                        int   ti = bi[t]; bi[t] = bi[t-1]; bi[t-1] = ti;
                    }
                }
            }
        }
        int* o = knn + (size_t)(b*S_ + q) * K_;
#pragma unroll
        for (int t = 0; t < K_; ++t) o[t] = bi[t];
    }
}

// ---------------------------------------------------------------------------
// 3) Pad w0 [128,67] -> [128,68] so the WMMA K-loop needs no guards.
// ---------------------------------------------------------------------------
__global__ void padw0_kernel(const float* __restrict__ w0, float* __restrict__ wp) {
    int e = blockIdx.x * 256 + threadIdx.x;
    if (e >= C_*KPAD0) return;
    int o = e / KPAD0, c = e % KPAD0;
    wp[e] = (c < CIN0) ? w0[o*CIN0 + c] : 0.0f;
}

// ---------------------------------------------------------------------------
// 4) Layer-0 GEMM with fused gather/concat. Block = 64 rows x 128 cols;
//    8 waves, each owns four 16x16 tiles (one B-fragment reused against 4
//    A-fragments per k-step -> 4x less W traffic, 4 independent accumulators
//    for WMMA ILP). V_WMMA_F32_16X16X4_F32 keeps full fp32 fidelity: at
//    ~0.3 GB of stage traffic vs 2.3 GFLOP, this GEMM is bandwidth-bound,
//    so low-precision matrix formats buy nothing here.
// ---------------------------------------------------------------------------
__global__ void __launch_bounds__(256)
gemm0_kernel(const float* __restrict__ xyz, const float* __restrict__ new_xyz,
             const float* __restrict__ feat, const int* __restrict__ knn,
             const float* __restrict__ wp, const float* __restrict__ bias,
             float* __restrict__ y) {
    __shared__ float As[64 * KPAD0];   // 17 KB
    const int mtile = blockIdx.x * 64;
    const int tid = threadIdx.x;
    for (int t = tid; t < 64*KPAD0; t += 256) {
        const int row = mtile + t / KPAD0;
        const int c   = t % KPAD0;
        const int b   = row / (S_*K_);
        const int rem = row % (S_*K_);
        const int s   = rem / K_;
        const int id  = knn[row];
        float v;
        if (c < 3)          v = xyz[((size_t)b*N_ + id)*3 + c]
                              - new_xyz[((size_t)b*S_ + s)*3 + c];
        else if (c < CIN0)  v = feat[((size_t)b*N_ + id)*F_ + (c-3)];
        else                v = 0.0f;
        As[t] = v;
    }
    __syncthreads();

    const int lane  = tid & 31;
    const int ntile = (tid >> 5) * 16;
    const int mrow  = lane & 15;
    const int kl    = (lane >> 4) * 2;
    v8f acc[4] = {};
    for (int k0 = 0; k0 < KPAD0; k0 += 4) {
        const float* wrow = wp + (ntile + mrow)*KPAD0 + k0 + kl;
        v2f bb; bb[0] = wrow[0]; bb[1] = wrow[1];
#pragma unroll
        for (int mt = 0; mt < 4; ++mt) {
            v2f a;
            a[0] = As[(mt*16 + mrow)*KPAD0 + k0 + kl];
            a[1] = As[(mt*16 + mrow)*KPAD0 + k0 + kl + 1];
            acc[mt] = __builtin_amdgcn_wmma_f32_16x16x4_f32(
                          false, a, false, bb, (short)0, acc[mt], false, false);
        }
    }
    const float bvv = bias[ntile + mrow];
    const int mh = (lane >> 4) * 8;
#pragma unroll
    for (int mt = 0; mt < 4; ++mt)
#pragma unroll
        for (int r = 0; r < 8; ++r)
            y[(size_t)(mtile + mt*16 + mh + r)*C_ + (ntile + mrow)] =
                acc[mt][r] + bvv;
}

// ---------------------------------------------------------------------------
// 5) Layer-1 GEMM: [M,128] x [128,128]^T. The A-strip is a regular 64x128
//    f32 tile -> stage it with the Tensor Data Mover (one wave builds the
//    D# descriptor, issues TENSOR_LOAD_TO_LDS, waits TENSORcnt, then the
//    workgroup barrier publishes the LDS tile). Arity hedged per toolchain;
//    manual-copy fallback if the builtin is absent.
// ---------------------------------------------------------------------------
__global__ void __launch_bounds__(256)
gemm1_kernel(const float* __restrict__ X, const float* __restrict__ w,
             const float* __restrict__ bias, float* __restrict__ y) {
    __shared__ float As[64 * C_];      // 32 KB; sole LDS object -> offset 0
    const int mtile = blockIdx.x * 64;
    const int tid = threadIdx.x;

#if __has_builtin(__builtin_amdgcn_tensor_load_to_lds) && \
    __has_builtin(__builtin_amdgcn_s_wait_tensorcnt)
    if (tid < 32) {                    // single wave issues the TDM op
        unsigned long long gaddr =
            (unsigned long long)(const void*)(X + (size_t)mtile * C_);
        // D# group 0: count=1 user descriptor, lds_addr=0, global_addr, type=2
        u32x4 g0 = {0u, 0u, 0u, 0u};
        g0[0] = 1u;
        g0[1] = 0u;                                    // LDS byte offset of As
        g0[2] = (unsigned)(gaddr & 0xffffffffull);
        g0[3] = (unsigned)((gaddr >> 32) & 0x01ffffffull) | (2u << 30);
        // D# group 1: data_size=4B; tensor 128 x 131072 (dim0 x dim1),
        // tile 128 x 64, dim0 stride = 128 elements; no multicast/pad/iterate.
        i32x8 g1 = {0, 0, 0, 0, 0, 0, 0, 0};
        g1[0] = (int)(2u << 16);                       // data_size = 4 bytes
        g1[1] = (int)(((unsigned)C_ & 0xffffu) << 16); // tensor_dim0 lo16
        g1[2] = (int)((((unsigned)C_ >> 16) & 0xffffu)
                      | (((unsigned)M_ & 0xffffu) << 16)); // dim0 hi | dim1 lo
        g1[3] = (int)((((unsigned)M_ >> 16) & 0xffffu)
                      | ((unsigned)C_ << 16));         // dim1 hi | tile_dim0
        g1[4] = 64;                                    // tile_dim1 (rows)
        g1[5] = (int)C_;                               // tensor_dim0_stride
        i32x4 g2 = {0, 0, 0, 0};                       // 2D tensor: groups 2/3
        i32x4 g3 = {0, 0, 0, 0};                       //   unused (NULL-like)
#if __clang_major__ >= 23
        i32x8 g4 = {0, 0, 0, 0, 0, 0, 0, 0};
        __builtin_amdgcn_tensor_load_to_lds(g0, g1, g2, g3, g4, 0);
#else
        __builtin_amdgcn_tensor_load_to_lds(g0, g1, g2, g3, 0);
#endif
        __builtin_amdgcn_s_wait_tensorcnt(0);
    }
    __syncthreads();
#else
    for (int t = tid; t < 64*C_; t += 256) As[t] = X[(size_t)mtile*C_ + t];
    __syncthreads();
#endif

    const int lane  = tid & 31;
    const int ntile = (tid >> 5) * 16;
    const int mrow  = lane & 15;
    const int kl    = (lane >> 4) * 2;
    v8f acc[4] = {};
    for (int k0 = 0; k0 < C_; k0 += 4) {
        const float* wrow = w + (ntile + mrow)*C_ + k0 + kl;
        v2f bb; bb[0] = wrow[0]; bb[1] = wrow[1];
#pragma unroll
        for (int mt = 0; mt < 4; ++mt) {
            v2f a;
            a[0] = As[(mt*16 + mrow)*C_ + k0 + kl];
            a[1] = As[(mt*16 + mrow)*C_ + k0 + kl + 1];
            acc[mt] = __builtin_amdgcn_wmma_f32_16x16x4_f32(
                          false, a, false, bb, (short)0, acc[mt], false, false);
        }
    }
    const float bvv = bias[ntile + mrow];
    const int mh = (lane >> 4) * 8;
#pragma unroll
    for (int mt = 0; mt < 4; ++mt)
#pragma unroll
        for (int r = 0; r < 8; ++r)
            y[(size_t)(mtile + mt*16 + mh + r)*C_ + (ntile + mrow)] =
                acc[mt][r] + bvv;
}

// ---------------------------------------------------------------------------
// 6) Deterministic per-channel BN stats: one block per channel, tree reduce.
// ---------------------------------------------------------------------------
__global__ void __launch_bounds__(256)
stats_kernel(const float* __restrict__ y, float* __restrict__ mean,
             float* __restrict__ var) {
    __shared__ float ssum[256], ssq[256];
    const int c = blockIdx.x, tid = threadIdx.x;
    float s = 0.f, q = 0.f;
    for (int r = tid; r < M_; r += 256) {
        float v = y[(size_t)r*C_ + c];
        s += v; q += v*v;
    }
    ssum[tid] = s; ssq[tid] = q;
    __syncthreads();
    for (int off = 128; off > 0; off >>= 1) {
        if (tid < off) { ssum[tid] += ssum[tid+off]; ssq[tid] += ssq[tid+off]; }
        __syncthreads();
    }
    if (tid == 0) {
        float m = ssum[0] / (float)M_;
        mean[c] = m;
        var[c]  = ssq[0] / (float)M_ - m*m;   // biased variance
    }
}

// 7) BN + ReLU in place (layer 0)
__global__ void bnrelu_kernel(float* __restrict__ y, const float* __restrict__ mean,
                              const float* __restrict__ var, const float* __restrict__ g,
                              const float* __restrict__ be) {
    size_t e = (size_t)blockIdx.x * 256 + threadIdx.x;
    if (e >= (size_t)M_ * C_) return;
    int c = (int)(e & (C_-1));
    float v = (y[e] - mean[c]) * rsqrtf(var[c] + BN_EPS) * g[c] + be[c];
    y[e] = fmaxf(v, 0.0f);
}

// 8) Fused BN + ReLU + max over K for layer 1 -> new_features [B,S,C]
__global__ void final_kernel(const float* __restrict__ y1, const float* __restrict__ mean,
                             const float* __restrict__ var, const float* __restrict__ g,
                             const float* __restrict__ be, float* __restrict__ out) {
    int e = blockIdx.x * 256 + threadIdx.x;   // B*S*C elements
    if (e >= B_*S_*C_) return;
    const int c  = e & (C_-1);
    const int bs = e >> 7;
    const float mn = mean[c];
    const float sc = rsqrtf(var[c] + BN_EPS) * g[c];
    const float bt = be[c];
    float mx = 0.0f;                          // ReLU outputs are >= 0
#pragma unroll
    for (int kk = 0; kk < K_; ++kk) {
        float v = y1[(size_t)(bs*K_ + kk)*C_ + c];
        v = fmaxf((v - mn)*sc + bt, 0.0f);
        mx = fmaxf(mx, v);
    }
    out[e] = mx;
}

// ---------------------------------------------------------------------------
extern "C" void kernel_launch(void* const* d_in, const int* in_sizes, int n_in,
                              void* d_out, int out_size, void* d_ws, size_t ws_size,
                              hipStream_t stream) {
    const float* xyz  = (const float*)d_in[0];
    const float* feat = (const float*)d_in[1];
    const float* w0   = (const float*)d_in[2];
    const float* b0   = (const float*)d_in[3];
    const float* g0   = (const float*)d_in[4];
    const float* be0  = (const float*)d_in[5];
    const float* w1   = (const float*)d_in[6];
    const float* b1   = (const float*)d_in[7];
    const float* g1   = (const float*)d_in[8];
    const float* be1  = (const float*)d_in[9];

    float* out      = (float*)d_out;
    float* new_xyz  = out;                    // [B,S,3]
    float* new_feat = out + B_*S_*3;          // [B,S,C]

    char* ws = (char*)d_ws;
    int*   knn   = (int*)ws;                                        // M_ ints
    float* y0    = (float*)(ws + 524288);                           // M_*C_
    float* y1    = (float*)(ws + 524288 + (size_t)M_*C_*4);         // M_*C_
    float* stats = (float*)(ws + 524288 + 2*(size_t)M_*C_*4);       // 512 f
    float* mean0 = stats,       *var0 = stats + 128;
    float* mean1 = stats + 256, *var1 = stats + 384;
    float* wp    = stats + 512;                                     // 128*68

    fps_kernel <<<B_, 1024, 0, stream>>>(xyz, new_xyz);
    knn_kernel <<<B_, 256, (N_*3 + S_*3)*sizeof(float), stream>>>(xyz, new_xyz, knn);
    padw0_kernel<<<(C_*KPAD0 + 255)/256, 256, 0, stream>>>(w0, wp);
    gemm0_kernel<<<M_/64, 256, 0, stream>>>(xyz, new_xyz, feat, knn, wp, b0, y0);
    stats_kernel<<<C_, 256, 0, stream>>>(y0, mean0, var0);
    bnrelu_kernel<<<(int)(((size_t)M_*C_ + 255)/256), 256, 0, stream>>>(y0, mean0, var0, g0, be0);
    gemm1_kernel<<<M_/64, 256, 0, stream>>>(y0, w1, b1, y1);
    stats_kernel<<<C_, 256, 0, stream>>>(y1, mean1, var1);
    final_kernel<<<(B_*S_*C_ + 255)/256, 256, 0, stream>>>(y1, mean1, var1, g1, be1, new_feat);
}